// VecInt_38104949850217
// MI455X (gfx1250) — compile-verified
//
#include <hip/hip_runtime.h>
#include <hip/hip_bf16.h>
#include <stdint.h>

// VecInt: scaling-and-squaring integration of a stationary velocity field.
//   v0 = vel / 2^7 ; repeat 7x: v <- v + trilinear_warp(v, v)
// Shapes: vel/out = [2,128,128,128,3] float32. Voxel count per batch = 2^21.
//
// Memory plan: ping-pong between d_ws (>= 50.3MB needed) and d_out so the
// 7th warp step writes d_out. Both buffers (100MB) are L2-resident (192MB L2).

#define DIM      128
#define DIM_M1   127
#define LOG2_D   7           // bits per spatial axis
#define VOX_BITS 21          // 128^3 = 2^21 voxels per batch
#define INT_STEPS 7

// ---------------------------------------------------------------------------
// Pass 1: v0 = vel * 2^-7, streamed through LDS via the CDNA5 async data-mover
// path (GLOBAL_LOAD_ASYNC_TO_LDS_B128, tracked by ASYNCcnt). Purely streaming,
// HBM-bound; the LDS staging costs nothing extra and lights up the gfx1250
// async path. Each lane reads back only the LDS bytes its own async op wrote,
// so s_wait_asynccnt 0 (per-wave) is sufficient — no barrier needed.
// ---------------------------------------------------------------------------
__global__ __launch_bounds__(256) void scale_async_kernel(
    const float4* __restrict__ src, float4* __restrict__ dst, int n4, float s)
{
    __shared__ float4 sbuf[256];
    int i = blockIdx.x * 256 + threadIdx.x;
    if (i >= n4) return;

    // Low 32 bits of a generic pointer into LDS are the DS byte offset
    // (flat-LDS aperture mapping).
    unsigned lds_off = (unsigned)(uintptr_t)(&sbuf[threadIdx.x]);
    unsigned long long gaddr = (unsigned long long)(uintptr_t)(src + i);

    asm volatile("global_load_async_to_lds_b128 %0, %1, off"
                 :: "v"(lds_off), "v"(gaddr) : "memory");
    asm volatile("s_wait_asynccnt 0" ::: "memory");

    float4 v = sbuf[threadIdx.x];
    float4 r;
    r.x = v.x * s; r.y = v.y * s; r.z = v.z * s; r.w = v.w * s;
    dst[i] = r;
}

// ---------------------------------------------------------------------------
// Pass 2..8: dst = src + trilinear_warp(src, src), border-clipped.
// One thread per voxel; 1 center b96 load + 8 gathered b96 corner loads
// + 1 b96 store. All intermediate passes hit in L2 (both buffers resident).
// ---------------------------------------------------------------------------
__device__ __forceinline__ void gather_acc(const float* __restrict__ p, float w,
                                           float& ox, float& oy, float& oz)
{
    ox = fmaf(w, p[0], ox);
    oy = fmaf(w, p[1], oy);
    oz = fmaf(w, p[2], oz);
}

__global__ __launch_bounds__(256) void warp_step_kernel(
    const float* __restrict__ src, float* __restrict__ dst, int nvox)
{
    int i = blockIdx.x * 256 + threadIdx.x;
    if (i >= nvox) return;

    const int W = i & DIM_M1;                 // fastest spatial axis (w)
    const int H = (i >> LOG2_D) & DIM_M1;     // h
    const int D = (i >> (2 * LOG2_D)) & DIM_M1; // d (slowest spatial)
    const int bbase = i & ~((1 << VOX_BITS) - 1); // batch * 128^3

    const float* c = src + (size_t)i * 3;
    float fx = c[0], fy = c[1], fz = c[2];    // flow: (d, h, w) displacement

    // loc = clip(grid + flow, 0, 127)
    float ld = fminf(fmaxf((float)D + fx, 0.0f), 127.0f);
    float lh = fminf(fmaxf((float)H + fy, 0.0f), 127.0f);
    float lw = fminf(fmaxf((float)W + fz, 0.0f), 127.0f);

    float fd = floorf(ld), fh = floorf(lh), fw = floorf(lw);
    float wd1 = ld - fd, wh1 = lh - fh, ww1 = lw - fw;
    float wd0 = 1.0f - wd1, wh0 = 1.0f - wh1, ww0 = 1.0f - ww1;

    int d0 = (int)fd, h0 = (int)fh, w0 = (int)fw;
    int d1 = min(d0 + 1, DIM_M1);
    int h1 = min(h0 + 1, DIM_M1);
    int w1 = min(w0 + 1, DIM_M1);

    int rd0 = bbase + (d0 << (2 * LOG2_D));
    int rd1 = bbase + (d1 << (2 * LOG2_D));
    int rh0 = h0 << LOG2_D;
    int rh1 = h1 << LOG2_D;

    float ox = 0.0f, oy = 0.0f, oz = 0.0f;
    // Corner order matches reference loop nest (cx->d, cy->h, cz->w).
    gather_acc(src + (size_t)(rd0 + rh0 + w0) * 3, wd0 * wh0 * ww0, ox, oy, oz);
    gather_acc(src + (size_t)(rd0 + rh0 + w1) * 3, wd0 * wh0 * ww1, ox, oy, oz);
    gather_acc(src + (size_t)(rd0 + rh1 + w0) * 3, wd0 * wh1 * ww0, ox, oy, oz);
    gather_acc(src + (size_t)(rd0 + rh1 + w1) * 3, wd0 * wh1 * ww1, ox, oy, oz);
    gather_acc(src + (size_t)(rd1 + rh0 + w0) * 3, wd1 * wh0 * ww0, ox, oy, oz);
    gather_acc(src + (size_t)(rd1 + rh0 + w1) * 3, wd1 * wh0 * ww1, ox, oy, oz);
    gather_acc(src + (size_t)(rd1 + rh1 + w0) * 3, wd1 * wh1 * ww0, ox, oy, oz);
    gather_acc(src + (size_t)(rd1 + rh1 + w1) * 3, wd1 * wh1 * ww1, ox, oy, oz);

    float* o = dst + (size_t)i * 3;
    o[0] = fx + ox;
    o[1] = fy + oy;
    o[2] = fz + oz;
}

// ---------------------------------------------------------------------------
extern "C" void kernel_launch(void* const* d_in, const int* in_sizes, int n_in,
                              void* d_out, int out_size, void* d_ws, size_t ws_size,
                              hipStream_t stream)
{
    (void)n_in; (void)out_size; (void)ws_size;

    const float* vel = (const float*)d_in[0];
    float* out = (float*)d_out;
    float* ws  = (float*)d_ws;       // needs >= in_sizes[0]*4 bytes (50.3 MB)

    const int total = in_sizes[0];   // 2*128^3*3 = 12,582,912 floats
    const int nvox  = total / 3;     // 4,194,304 voxels
    const int n4    = total / 4;     // float4 count (total % 4 == 0)
    const float s   = 1.0f / (float)(1 << INT_STEPS);

    // Pass 1: ws = vel * 2^-7  (async-to-LDS staged, streaming)
    scale_async_kernel<<<(n4 + 255) / 256, 256, 0, stream>>>(
        (const float4*)vel, (float4*)ws, n4, s);

    // Passes 2..8: 7 squaring steps, ping-pong ws <-> out.
    // Sequence of write targets: ws, out, ws, out, ws, out, ws, out
    // -> step 7 lands in d_out.
    float* cur = ws;
    for (int t = 0; t < INT_STEPS; ++t) {
        float* dst = (cur == ws) ? out : ws;
        warp_step_kernel<<<(nvox + 255) / 256, 256, 0, stream>>>(cur, dst, nvox);
        cur = dst;
    }
}